// TrafficGNN_72086731096216
// MI455X (gfx1250) — compile-verified
//
#include <hip/hip_runtime.h>
#include <hip/hip_bf16.h>

typedef __attribute__((ext_vector_type(2))) float v2f;
typedef __attribute__((ext_vector_type(8))) float v8f;

// ---------------------------------------------------------------------------
// Phase 1: degree + normalization
// ---------------------------------------------------------------------------
__global__ void k_init_deg(float* __restrict__ deg, int n) {
  int i = blockIdx.x * blockDim.x + threadIdx.x;
  if (i < n) deg[i] = 1.0f;  // self-loop contribution
}

__global__ void k_count_deg(const int* __restrict__ col, float* __restrict__ deg, int e) {
  int i = blockIdx.x * blockDim.x + threadIdx.x;
  if (i < e) atomicAdd(&deg[col[i]], 1.0f);
}

__global__ void k_rsqrt_deg(float* __restrict__ deg, int n) {
  int i = blockIdx.x * blockDim.x + threadIdx.x;
  if (i < n) deg[i] = rsqrtf(deg[i]);  // deg >= 1 always (self loop)
}

// ---------------------------------------------------------------------------
// Phase 2: h1 = x @ W1   via V_WMMA_F32_16X16X4_F32
// One wave computes a 16-node x 16-feature tile. K=5 padded to 8 -> 2 WMMAs.
// A layout (32-bit A 16x4): lane<16 -> K={k0,k0+1}, lane>=16 -> K={k0+2,k0+3},
//   M = lane&15. B layout mirrors (N = lane&15). C/D: VGPR j -> M=j+8*half.
// Full-tile fast path keeps the store loop branch-free (wave-uniform test).
// ---------------------------------------------------------------------------
__global__ void k_gemm1_wmma(const float* __restrict__ x,
                             const float* __restrict__ W1,
                             float* __restrict__ h1, int nTiles, int n) {
  int wave = (int)((blockIdx.x * (unsigned)blockDim.x + threadIdx.x) >> 5);
  if (wave >= nTiles) return;            // wave-uniform: EXEC stays all-ones
  int lane = threadIdx.x & 31;
  int nn   = lane & 15;                  // M index for A, N index for B
  int half = lane >> 4;                  // 0 or 1
  int base = wave * 16;
  bool full = (base + 16 <= n);          // wave-uniform

  int row  = base + nn;
  int rowc = full ? row : (row < n ? row : (n - 1));
  const float* xr = x + (long)rowc * 5;
  int k0 = half * 2;                     // {0,2}

  v2f a0, a1, b0, b1;
  a0.x = xr[k0];
  a0.y = xr[k0 + 1];
  a1.x = (half == 0) ? xr[4] : 0.0f;     // K=4 valid, K=5..7 padded zero
  a1.y = 0.0f;

  b0.x = W1[k0 * 16 + nn];
  b0.y = W1[(k0 + 1) * 16 + nn];
  b1.x = (half == 0) ? W1[4 * 16 + nn] : 0.0f;
  b1.y = 0.0f;

  v8f c = {0.f, 0.f, 0.f, 0.f, 0.f, 0.f, 0.f, 0.f};
  c = __builtin_amdgcn_wmma_f32_16x16x4_f32(false, a0, false, b0, (short)0, c, false, false);
  c = __builtin_amdgcn_wmma_f32_16x16x4_f32(false, a1, false, b1, (short)0, c, false, false);

  // Lane's 8 results live at rows base+8*half+j, column nn.
  float* p = h1 + (long)(base + 8 * half) * 16 + nn;
  if (full) {
#pragma unroll
    for (int j = 0; j < 8; ++j) p[j * 16] = c[j];
  } else {
#pragma unroll
    for (int j = 0; j < 8; ++j) {
      if (base + 8 * half + j < n) p[j * 16] = c[j];
    }
  }
}

// ---------------------------------------------------------------------------
// Phase 3: aggregation of layer 1 (self-loop folded into the init)
// ---------------------------------------------------------------------------
__global__ void k_init_agg(const float* __restrict__ dinv,
                           const float* __restrict__ h1,
                           float* __restrict__ agg, int n) {
  int i = blockIdx.x * blockDim.x + threadIdx.x;
  if (i >= n) return;
  float w = dinv[i] * dinv[i];
  const float4* src = (const float4*)(h1 + (long)i * 16);
  float4* dst = (float4*)(agg + (long)i * 16);
#pragma unroll
  for (int j = 0; j < 4; ++j) {
    float4 v = src[j];
    v.x *= w; v.y *= w; v.z *= w; v.w *= w;
    dst[j] = v;
  }
}

__global__ void k_scatter1(const int* __restrict__ row,
                           const int* __restrict__ col,
                           const float* __restrict__ dinv,
                           const float* __restrict__ h1,
                           float* __restrict__ agg, int e) {
  int i = blockIdx.x * blockDim.x + threadIdx.x;
  if (i >= e) return;
  int r = row[i], c = col[i];
  float w = dinv[r] * dinv[c];
  const float4* src = (const float4*)(h1 + (long)r * 16);
  float* dst = agg + (long)c * 16;
#pragma unroll
  for (int j = 0; j < 4; ++j) {
    float4 v = src[j];
    atomicAdd(dst + 4 * j + 0, w * v.x);
    atomicAdd(dst + 4 * j + 1, w * v.y);
    atomicAdd(dst + 4 * j + 2, w * v.z);
    atomicAdd(dst + 4 * j + 3, w * v.w);
  }
}

// ---------------------------------------------------------------------------
// Phase 4: bias + relu + W2 dot, then layer-2 self-loop + b2 into out
// ---------------------------------------------------------------------------
__global__ void k_post1(const float* __restrict__ agg,
                        const float* __restrict__ b1,
                        const float* __restrict__ W2,
                        const float* __restrict__ b2,
                        const float* __restrict__ dinv,
                        float* __restrict__ h2,
                        float* __restrict__ out, int n) {
  int i = blockIdx.x * blockDim.x + threadIdx.x;
  if (i >= n) return;
  float acc = 0.0f;
#pragma unroll
  for (int f = 0; f < 16; ++f) {
    float v = agg[(long)i * 16 + f] + b1[f];
    v = fmaxf(v, 0.0f);
    acc = fmaf(v, W2[f], acc);
  }
  h2[i] = acc;
  out[i] = dinv[i] * dinv[i] * acc + b2[0];
}

__global__ void k_scatter2(const int* __restrict__ row,
                           const int* __restrict__ col,
                           const float* __restrict__ dinv,
                           const float* __restrict__ h2,
                           float* __restrict__ out, int e) {
  int i = blockIdx.x * blockDim.x + threadIdx.x;
  if (i >= e) return;
  int r = row[i], c = col[i];
  atomicAdd(&out[c], dinv[r] * dinv[c] * h2[r]);
}

// ---------------------------------------------------------------------------
extern "C" void kernel_launch(void* const* d_in, const int* in_sizes, int n_in,
                              void* d_out, int out_size, void* d_ws, size_t ws_size,
                              hipStream_t stream) {
  const float* x  = (const float*)d_in[0];
  const int*   ei = (const int*)d_in[1];
  const float* W1 = (const float*)d_in[2];
  const float* b1 = (const float*)d_in[3];
  const float* W2 = (const float*)d_in[4];
  const float* b2 = (const float*)d_in[5];
  float* out = (float*)d_out;

  const int n = in_sizes[0] / 5;   // 100000 nodes
  const int e = in_sizes[1] / 2;   // 3200000 edges
  const int* row = ei;             // edge_index[0] = sources
  const int* col = ei + e;         // edge_index[1] = targets

  // Workspace layout (floats): dinv | h1[n*16] | agg[n*16] | h2[n]
  float* dinv = (float*)d_ws;
  float* h1   = dinv + (size_t)((n + 63) & ~63);
  float* agg  = h1 + (size_t)n * 16;
  float* h2   = agg + (size_t)n * 16;

  const int T = 256;
  const int gN = (n + T - 1) / T;
  const int gE = (e + T - 1) / T;

  k_init_deg <<<gN, T, 0, stream>>>(dinv, n);
  k_count_deg<<<gE, T, 0, stream>>>(col, dinv, e);
  k_rsqrt_deg<<<gN, T, 0, stream>>>(dinv, n);

  const int nTiles = (n + 15) / 16;          // 6250 waves, 8 waves / block
  const int gW = (nTiles * 32 + T - 1) / T;
  k_gemm1_wmma<<<gW, T, 0, stream>>>(x, W1, h1, nTiles, n);

  k_init_agg <<<gN, T, 0, stream>>>(dinv, h1, agg, n);
  k_scatter1 <<<gE, T, 0, stream>>>(row, col, dinv, h1, agg, e);
  k_post1    <<<gN, T, 0, stream>>>(agg, b1, W2, b2, dinv, h2, out, n);
  k_scatter2 <<<gE, T, 0, stream>>>(row, col, dinv, h2, out, e);
}